// CAM_Module_1520418422921
// MI455X (gfx1250) — compile-verified
//
#include <hip/hip_runtime.h>

typedef float v2f __attribute__((ext_vector_type(2)));
typedef float v8f __attribute__((ext_vector_type(8)));

#define BATCH 128
#define CCH   3
#define NTOT  200704          // 16*112*112
#define N4    (NTOT / 4)      // 50176 float4 per channel

// ---- kernel 1: gram partials ------------------------------------------------
#define S1     14             // splits per batch: 200704/14 = 14336 exactly
#define SLICE4 (NTOT / S1 / 4)// 3584 float4 per slice
#define T1     256
#define IT1    (SLICE4 / T1)  // 14 iterations exactly

__global__ __launch_bounds__(T1)
void gram_partial(const float* __restrict__ x, float* __restrict__ gpart) {
    const int blk = blockIdx.x;
    const int b = blk / S1, s = blk % S1;
    const float4* __restrict__ x0 = (const float4*)(x + (size_t)b * CCH * NTOT);
    const float4* __restrict__ x1 = x0 + N4;
    const float4* __restrict__ x2 = x0 + 2 * N4;

    const int f0 = s * SLICE4 + threadIdx.x;
    float s00 = 0.f, s01 = 0.f, s02 = 0.f, s11 = 0.f, s12 = 0.f, s22 = 0.f;
#pragma unroll
    for (int i = 0; i < IT1; ++i) {
        const int f = f0 + i * T1;
        const float4 a = x0[f];
        const float4 c = x1[f];
        const float4 e = x2[f];
        s00 += a.x*a.x + a.y*a.y + a.z*a.z + a.w*a.w;
        s01 += a.x*c.x + a.y*c.y + a.z*c.z + a.w*c.w;
        s02 += a.x*e.x + a.y*e.y + a.z*e.z + a.w*e.w;
        s11 += c.x*c.x + c.y*c.y + c.z*c.z + c.w*c.w;
        s12 += c.x*e.x + c.y*e.y + c.z*e.z + c.w*e.w;
        s22 += e.x*e.x + e.y*e.y + e.z*e.z + e.w*e.w;
    }
    // wave32 reduction
#pragma unroll
    for (int off = 16; off > 0; off >>= 1) {
        s00 += __shfl_down(s00, off, 32);
        s01 += __shfl_down(s01, off, 32);
        s02 += __shfl_down(s02, off, 32);
        s11 += __shfl_down(s11, off, 32);
        s12 += __shfl_down(s12, off, 32);
        s22 += __shfl_down(s22, off, 32);
    }
    __shared__ float red[T1 / 32][6];
    const int lane = threadIdx.x & 31, w = threadIdx.x >> 5;
    if (lane == 0) {
        red[w][0] = s00; red[w][1] = s01; red[w][2] = s02;
        red[w][3] = s11; red[w][4] = s12; red[w][5] = s22;
    }
    __syncthreads();
    if (threadIdx.x < 6) {
        float t = 0.f;
#pragma unroll
        for (int j = 0; j < T1 / 32; ++j) t += red[j][threadIdx.x];
        gpart[(size_t)blk * 8 + threadIdx.x] = t;
    }
}

// ---- kernel 2: 3x3 softmax attention, gamma folded in -----------------------
__global__ void attention_k(const float* __restrict__ gpart,
                            const float* __restrict__ gamma,
                            float* __restrict__ att) {
    const int b = blockIdx.x * blockDim.x + threadIdx.x;
    if (b >= BATCH) return;
    float g[6] = {0.f, 0.f, 0.f, 0.f, 0.f, 0.f};
    for (int s = 0; s < S1; ++s)
#pragma unroll
        for (int j = 0; j < 6; ++j) g[j] += gpart[(size_t)(b * S1 + s) * 8 + j];

    const float e[3][3] = {{g[0], g[1], g[2]},
                           {g[1], g[3], g[4]},
                           {g[2], g[4], g[5]}};
    const float ga = gamma[0];
#pragma unroll
    for (int c = 0; c < 3; ++c) {
        const float rmx = fmaxf(e[c][0], fmaxf(e[c][1], e[c][2]));
        const float en0 = rmx - e[c][0], en1 = rmx - e[c][1], en2 = rmx - e[c][2];
        const float smx = fmaxf(en0, fmaxf(en1, en2));
        const float x0 = expf(en0 - smx), x1 = expf(en1 - smx), x2 = expf(en2 - smx);
        const float inv = ga / (x0 + x1 + x2);
        att[b * 12 + c * 3 + 0] = x0 * inv;
        att[b * 12 + c * 3 + 1] = x1 * inv;
        att[b * 12 + c * 3 + 2] = x2 * inv;
    }
}

// ---- kernel 3: out = att@x + x via v_wmma_f32_16x16x4_f32 -------------------
#define CHUNK_N 2048                  // 8 waves * 16 tiles * 16 n
#define CH_PER_B (NTOT / CHUNK_N)     // 98 exactly

__global__ __launch_bounds__(256)
void apply_k(const float* __restrict__ x, const float* __restrict__ att,
             float* __restrict__ out) {
    // reversed order vs kernel 1 -> re-read the L2-resident tail of x first
    const int blk = (int)gridDim.x - 1 - (int)blockIdx.x;
    const int b = blk / CH_PER_B, ch = blk % CH_PER_B;
    const float* __restrict__ xb = x + (size_t)b * CCH * NTOT;
    float* __restrict__ ob = out + (size_t)b * CCH * NTOT;
    const float* __restrict__ A9 = att + b * 12;

    const int lane = threadIdx.x & 31;
    const int w    = threadIdx.x >> 5;
    const int m    = lane & 15;
    const bool lo  = lane < 16;

    // A operand (16x4): lanes 0-15 hold K=0,1 ; lanes 16-31 hold K=2,3
    v2f Aop;
    Aop.x = (m < 3) ? A9[m * 3 + (lo ? 0 : 2)] : 0.0f;   // K=0 / K=2
    Aop.y = (m < 3 && lo) ? A9[m * 3 + 1] : 0.0f;        // K=1 / K=3(=0 pad)

    // branchless load offsets: hi lanes reload x[0,n] for b1 (L0 hit, zeroed below)
    const int d0off = lo ? 0 : 2 * NTOT;
    const int b1off = lo ? NTOT : 0;

    const v8f Cz = {0.f, 0.f, 0.f, 0.f, 0.f, 0.f, 0.f, 0.f};
    const int nbase = ch * CHUNK_N + w * 256 + m;

#pragma unroll
    for (int g = 0; g < 4; ++g) {            // 4 groups of 4 tiles
        float p0[4], p1[4], q2[4];
#pragma unroll
        for (int j = 0; j < 4; ++j) {        // 8 unconditional loads, clause-able
            const int n = nbase + (g * 4 + j) * 16;
            p0[j] = xb[d0off + n];           // B K=0 row / K=2 row
            p1[j] = xb[b1off + n];           // B K=1 row / (pad, zeroed)
        }
#pragma unroll
        for (int j = 0; j < 4; ++j)          // x[2,n] into lanes 0-15
            q2[j] = __shfl(p0[j], 16 + m, 32);
#pragma unroll
        for (int j = 0; j < 4; ++j) {
            const int n = nbase + (g * 4 + j) * 16;
            v2f Bop;
            Bop.x = p0[j];
            Bop.y = lo ? p1[j] : 0.0f;       // v_cndmask, no EXEC branch
            v8f D = __builtin_amdgcn_wmma_f32_16x16x4_f32(
                /*neg_a=*/false, Aop, /*neg_b=*/false, Bop,
                /*c_mod=*/(short)0, Cz, /*reuse_a=*/false, /*reuse_b=*/false);
            if (lo) {                        // residual add + NT streaming stores
                __builtin_nontemporal_store(D[0] + p0[j], &ob[n]);
                __builtin_nontemporal_store(D[1] + p1[j], &ob[NTOT + n]);
                __builtin_nontemporal_store(D[2] + q2[j], &ob[2 * NTOT + n]);
            }
        }
    }
}

// ---- launcher ---------------------------------------------------------------
extern "C" void kernel_launch(void* const* d_in, const int* in_sizes, int n_in,
                              void* d_out, int out_size, void* d_ws, size_t ws_size,
                              hipStream_t stream) {
    (void)in_sizes; (void)n_in; (void)out_size; (void)ws_size;
    const float* x     = (const float*)d_in[0];
    const float* gamma = (const float*)d_in[1];
    float* out = (float*)d_out;

    float* gpart = (float*)d_ws;                     // 128*14*8 floats
    float* att   = gpart + (size_t)BATCH * S1 * 8;   // 128*12 floats

    gram_partial<<<dim3(BATCH * S1), dim3(T1), 0, stream>>>(x, gpart);
    attention_k<<<dim3(1), dim3(BATCH), 0, stream>>>(gpart, gamma, att);
    apply_k<<<dim3(BATCH * CH_PER_B), dim3(256), 0, stream>>>(x, att, out);
}